// ScatterAttention_33663953666223
// MI455X (gfx1250) — compile-verified
//
#include <hip/hip_runtime.h>
#include <hip/hip_bf16.h>

#define E_CNT 640000
#define N_SEG 50000
#define VDIM  128
#define KDIM  128

typedef __attribute__((ext_vector_type(2))) float v2f;
typedef __attribute__((ext_vector_type(8))) float v8f;

// Order-preserving float -> uint encoding for atomicMax-based segment max.
__device__ __forceinline__ unsigned enc_ord(float f) {
    unsigned u = __float_as_uint(f);
    return (u & 0x80000000u) ? ~u : (u | 0x80000000u);
}
__device__ __forceinline__ float dec_ord(unsigned u) {
    return (u & 0x80000000u) ? __uint_as_float(u ^ 0x80000000u)
                             : __uint_as_float(~u);
}

// ---------------------------------------------------------------------------
// Kernel 0: init. Zero attn output region (harness poisons d_out), seed
// segment max (encoded -inf) and denom accumulators.
// ---------------------------------------------------------------------------
__global__ void sa_init(float* __restrict__ attn_out,
                        unsigned* __restrict__ segmax,
                        float* __restrict__ denom) {
    int i = blockIdx.x * 256 + threadIdx.x;
    if (i < N_SEG * VDIM) attn_out[i] = 0.0f;
    if (i < N_SEG) { segmax[i] = 0x007FFFFFu; /* enc(-inf) */ denom[i] = 0.0f; }
}

// ---------------------------------------------------------------------------
// Kernel 1: keys_proj = attn_keys (N x K) * W^T (K x V) + b  via f32 WMMA.
// One wave per 16x16 output tile, K stepped by 4 (V_WMMA_F32_16X16X4_F32).
// A layout (16x4 f32): lane = 16*hi + m; VGPR0 = A[m, k+2*hi], VGPR1 = A[m, k+1+2*hi]
// B layout mirrors (rows striped across lanes): lane holds B[k+2*hi .. +1, n].
// B[k, v] = W[v, k], so both A and B per-lane fetches are contiguous float2.
// ---------------------------------------------------------------------------
__global__ void sa_gemm_wmma(const float* __restrict__ attn_keys,
                             const float* __restrict__ W,
                             const float* __restrict__ bias,
                             float* __restrict__ kp) {
    const int wave = threadIdx.x >> 5;
    const int lane = threadIdx.x & 31;
    const int tile = blockIdx.x * 4 + wave;      // 25000 tiles total
    const int rt = tile >> 3;                    // 0..3124 (row tiles of 16)
    const int ct = tile & 7;                     // 0..7    (col tiles of 16)
    const int row0 = rt * 16, col0 = ct * 16;
    const int l15 = lane & 15, hi = lane >> 4;

    const float* arow = attn_keys + (size_t)(row0 + l15) * KDIM + 2 * hi;
    const float* brow = W        + (size_t)(col0 + l15) * KDIM + 2 * hi;

    v8f c = {};
#pragma unroll
    for (int k = 0; k < KDIM; k += 4) {
        v2f a = *(const v2f*)(arow + k);   // 8B-aligned b64 load
        v2f b = *(const v2f*)(brow + k);
        c = __builtin_amdgcn_wmma_f32_16x16x4_f32(
                /*neg_a=*/false, a, /*neg_b=*/false, b,
                /*c_mod=*/(short)0, c, /*reuse_a=*/false, /*reuse_b=*/false);
    }

    const float bv = bias[col0 + l15];
#pragma unroll
    for (int r = 0; r < 8; ++r) {
        const int row = row0 + r + 8 * hi;       // C/D: VGPR r -> M=r (+8 for hi half)
        kp[(size_t)row * VDIM + col0 + l15] = c[r] + bv;
    }
}

// ---------------------------------------------------------------------------
// Kernel 2: per-edge dot product + segment max. One wave per edge:
// 32 lanes x float4 = 128 floats, coalesced b128 loads, wave32 xor-reduce.
// ---------------------------------------------------------------------------
__global__ void sa_dot(const float* __restrict__ sv,
                       const int* __restrict__ idx,
                       const float* __restrict__ kp,
                       float* __restrict__ probs,
                       unsigned* __restrict__ segmax) {
    const int wave = threadIdx.x >> 5;
    const int lane = threadIdx.x & 31;
    const int e = blockIdx.x * 8 + wave;
    const int seg = idx[e];

    const float4 a = ((const float4*)(sv + (size_t)e * VDIM))[lane];
    const float4 b = ((const float4*)(kp + (size_t)seg * VDIM))[lane];
    float p = a.x * b.x + a.y * b.y + a.z * b.z + a.w * b.w;
#pragma unroll
    for (int m = 16; m > 0; m >>= 1) p += __shfl_xor(p, m, 32);

    if (lane == 0) {
        probs[e] = p;
        atomicMax(&segmax[seg], enc_ord(p));
    }
}

// ---------------------------------------------------------------------------
// Kernel 3: ex = exp(probs - segmax[idx]); denom += ex. One thread per edge.
// ---------------------------------------------------------------------------
__global__ void sa_exp(const int* __restrict__ idx,
                       const unsigned* __restrict__ segmax,
                       float* __restrict__ probs_ex,
                       float* __restrict__ denom) {
    const int e = blockIdx.x * 256 + threadIdx.x;   // E divisible by 256
    const int seg = idx[e];
    const float ex = __expf(probs_ex[e] - dec_ord(segmax[seg]));
    probs_ex[e] = ex;
    atomicAdd(&denom[seg], ex);
}

// ---------------------------------------------------------------------------
// Kernel 4: scores = ex/denom[idx]; attn_out[idx,:] += sv[e,:]*score.
// One wave per edge, float4 per lane, f32 L2 atomics for the scatter-sum.
// ---------------------------------------------------------------------------
__global__ void sa_final(const float* __restrict__ sv,
                         const int* __restrict__ idx,
                         const float* __restrict__ ex,
                         const float* __restrict__ denom,
                         float* __restrict__ scores_out,
                         float* __restrict__ attn_out) {
    const int wave = threadIdx.x >> 5;
    const int lane = threadIdx.x & 31;
    const int e = blockIdx.x * 8 + wave;
    const int seg = idx[e];

    const float score = ex[e] / denom[seg];
    if (lane == 0) scores_out[e] = score;

    const float4 a = ((const float4*)(sv + (size_t)e * VDIM))[lane];
    float* dst = attn_out + (size_t)seg * VDIM + lane * 4;
    atomicAdd(dst + 0, a.x * score);
    atomicAdd(dst + 1, a.y * score);
    atomicAdd(dst + 2, a.z * score);
    atomicAdd(dst + 3, a.w * score);
}

// ---------------------------------------------------------------------------
extern "C" void kernel_launch(void* const* d_in, const int* in_sizes, int n_in,
                              void* d_out, int out_size, void* d_ws, size_t ws_size,
                              hipStream_t stream) {
    const float* sv        = (const float*)d_in[0];  // [E, V]
    const int*   idx       = (const int*)  d_in[1];  // [E]
    const float* attn_keys = (const float*)d_in[2];  // [N, K]
    const float* W         = (const float*)d_in[3];  // [V, K]
    const float* bias      = (const float*)d_in[4];  // [V]

    float* scores_out = (float*)d_out;               // [E]
    float* attn_out   = (float*)d_out + E_CNT;       // [N, V]

    // Workspace layout (floats): kp[N*V] | probs/ex[E] | segmax[N] (u32) | denom[N]
    float*    kp     = (float*)d_ws;
    float*    ex     = kp + (size_t)N_SEG * VDIM;
    unsigned* segmax = (unsigned*)(ex + E_CNT);
    float*    denom  = (float*)(segmax + N_SEG);

    // 0) init outputs + accumulators
    sa_init<<<(N_SEG * VDIM + 255) / 256, 256, 0, stream>>>(attn_out, segmax, denom);

    // 1) keys projection GEMM (WMMA f32): 25000 tiles, 4 waves/block
    sa_gemm_wmma<<<25000 / 4, 128, 0, stream>>>(attn_keys, W, bias, kp);

    // 2) per-edge logits + segment max: one wave per edge, 8 waves/block
    sa_dot<<<E_CNT / 8, 256, 0, stream>>>(sv, idx, kp, ex, segmax);

    // 3) exponentials + segment denominators
    sa_exp<<<E_CNT / 256, 256, 0, stream>>>(idx, segmax, ex, denom);

    // 4) normalize + weighted scatter aggregation
    sa_final<<<E_CNT / 8, 256, 0, stream>>>(sv, idx, ex, denom, scores_out, attn_out);
}